// BNetPred_64450279244061
// MI455X (gfx1250) — compile-verified
//
#include <hip/hip_runtime.h>
#include <hip/hip_bf16.h>
#include <math.h>

// Problem constants (match reference)
#define DD   64
#define GDIM 2016
#define NB   8192
#define HH   128
#define NOUT 128
#define ST   66   // padded LDS row stride (floats) to avoid bank conflicts

typedef __attribute__((ext_vector_type(2))) float v2f;
typedef __attribute__((ext_vector_type(8))) float v8f;

// ---------------------------------------------------------------------------
// One 16x16 output tile of a plain 64x64x64 f32 matmul from LDS:
//   Ctile = A[m0:m0+16, :] @ (TB ? B^T : B)[:, n0:n0+16]
// Pure ds_load + V_WMMA_F32_16X16X4_F32 (16 chained k-steps).
// Fragment layouts per ISA §7.12.2 (A: row-per-lane; B: col-per-lane;
// lanes 16..31 carry the K+2/K+3 pair).
// ---------------------------------------------------------------------------
template <bool TB>
__device__ __forceinline__ v8f mm64(const float* __restrict__ A,
                                    const float* __restrict__ B,
                                    int m0, int n0, int lane)
{
    v8f acc = {};
    const int lr   = lane & 15;
    const int hi   = lane >> 4;
    const int arow = (m0 + lr) * ST;
    const int gcol = n0 + lr;
#pragma unroll
    for (int k = 0; k < DD; k += 4) {
        const int ka = k + 2 * hi;
        v2f a, b;
        a.x = A[arow + ka];
        a.y = A[arow + ka + 1];
        if (TB) { b.x = B[gcol * ST + ka];  b.y = B[gcol * ST + ka + 1]; }
        else    { b.x = B[ka * ST + gcol];  b.y = B[(ka + 1) * ST + gcol]; }
        acc = __builtin_amdgcn_wmma_f32_16x16x4_f32(
            false, a, false, b, (short)0, acc, false, false);
    }
    return acc;
}

// Dst = C (plain)
__device__ __forceinline__ void store_tile(float* __restrict__ Dst, v8f c,
                                           int m0, int n0, int lane)
{
    const int lr = lane & 15;
    const int hi = lane >> 4;
#pragma unroll
    for (int r = 0; r < 8; ++r)
        Dst[(m0 + r + 8 * hi) * ST + (n0 + lr)] = c[r];
}

// Dst = alpha*I - C   (the aI-minus fold done once at store time, so every
// subsequent B-fragment read of Dst is a pure ds_load)
__device__ __forceinline__ void store_tile_aImC(float* __restrict__ Dst, v8f c,
                                                float alpha, int m0, int n0,
                                                int lane)
{
    const int lr  = lane & 15;
    const int hi  = lane >> 4;
    const int col = n0 + lr;
#pragma unroll
    for (int r = 0; r < 8; ++r) {
        const int row = m0 + r + 8 * hi;
        Dst[row * ST + col] = (row == col ? alpha : 0.0f) - c[r];
    }
}

// ---------------------------------------------------------------------------
// Kernel 1: per-sample Cayley map via 10-step order-7 Newton-Schulz.
// One workgroup (16 waves) per sample; all 64x64 operands resident in LDS.
// Outputs: R[n] (64x64, row-major) and mis[n] = R^T @ li.
// ---------------------------------------------------------------------------
__global__ __launch_bounds__(512) void cayley_ns_kernel(
    const float* __restrict__ gi, const float* __restrict__ li,
    const float* __restrict__ Wup, const float* __restrict__ bup,
    float* __restrict__ Rout, float* __restrict__ misout)
{
    __shared__ float sB[DD * ST];   // B = I - Askew   (note: Q = I + Askew = B^T)
    __shared__ float sZ[DD * ST];   // Z iterate
    __shared__ float sT1[DD * ST];  // BZ
    __shared__ float sT2[DD * ST];  // 7I - BZ, then 13I - U
    __shared__ float sT3[DD * ST];  // 15I - W
    __shared__ float sgi[DD], sli[DD];
    __shared__ float wsum[16], wmax[16];
    __shared__ float sinv;

    const int n    = blockIdx.x;
    const int tid  = threadIdx.x;
    const int lane = tid & 31;
    const int wave = tid >> 5;
    const int m0   = (wave >> 2) * 16;
    const int n0   = (wave & 3) * 16;

    if (tid < DD) { sgi[tid] = gi[n * DD + tid]; sli[tid] = li[n * DD + tid]; }
    __syncthreads();

    // vec = gi @ Wup^T + bup, scattered straight into skew form.
    // A[r][c] = vec_g (r>c);  Askew = A^T - A;  B = I - Askew.
    for (int g = tid; g < GDIM; g += 512) {
        float acc = bup[g];
        const float4* w4 = (const float4*)(Wup + (size_t)g * DD);
#pragma unroll
        for (int d = 0; d < DD / 4; ++d) {
            float4 wv = w4[d];
            acc += sgi[4 * d + 0] * wv.x + sgi[4 * d + 1] * wv.y +
                   sgi[4 * d + 2] * wv.z + sgi[4 * d + 3] * wv.w;
        }
        int r = (int)((1.0f + sqrtf(8.0f * (float)g + 1.0f)) * 0.5f);
        while (r * (r - 1) / 2 > g) --r;
        while ((r + 1) * r / 2 <= g) ++r;
        int c = g - r * (r - 1) / 2;
        sB[c * ST + r] = -acc;   // B[c][r] = -vec
        sB[r * ST + c] =  acc;   // B[r][c] = +vec
    }
    if (tid < DD) sB[tid * ST + tid] = 1.0f;
    __syncthreads();

    // norms = sum|B| * max|B|  ;  Z = B^T / norms
    float psum = 0.0f, pmax = 0.0f;
    for (int i = tid; i < DD * DD; i += 512) {
        float v = fabsf(sB[(i >> 6) * ST + (i & 63)]);
        psum += v; pmax = fmaxf(pmax, v);
    }
#pragma unroll
    for (int off = 16; off; off >>= 1) {
        psum += __shfl_down(psum, off, 32);
        pmax  = fmaxf(pmax, __shfl_down(pmax, off, 32));
    }
    if (lane == 0) { wsum[wave] = psum; wmax[wave] = pmax; }
    __syncthreads();
    if (tid == 0) {
        float s = 0.0f, m = 0.0f;
        for (int i = 0; i < 16; ++i) { s += wsum[i]; m = fmaxf(m, wmax[i]); }
        sinv = 1.0f / (s * m);
    }
    __syncthreads();
    {
        const float inv = sinv;
        for (int i = tid; i < DD * DD; i += 512) {
            int r = i >> 6, c = i & 63;
            sZ[r * ST + c] = sB[c * ST + r] * inv;
        }
    }
    __syncthreads();

    // 10x:  BZ = B@Z;  Z = 0.25 * Z @ (13I - BZ@(15I - BZ@(7I - BZ)))
    // All aI-X transforms applied at store time; loads stay pure.
    for (int it = 0; it < 10; ++it) {
        v8f c;
        // T1 = B@Z ; T2 = 7I - T1   (sT1/sT2 free: last read before prev barriers)
        c = mm64<false>(sB, sZ, m0, n0, lane);
        store_tile(sT1, c, m0, n0, lane);
        store_tile_aImC(sT2, c, 7.0f, m0, n0, lane);
        __syncthreads();
        // W = T1 @ T2 ; T3 = 15I - W
        c = mm64<false>(sT1, sT2, m0, n0, lane);
        store_tile_aImC(sT3, c, 15.0f, m0, n0, lane);
        __syncthreads();
        // U = T1 @ T3 ; T2 = 13I - U   (sT2 reads all finished at prev barrier)
        c = mm64<false>(sT1, sT3, m0, n0, lane);
        store_tile_aImC(sT2, c, 13.0f, m0, n0, lane);
        __syncthreads();
        // Z = 0.25 * Z @ T2   (overwrites sZ -> barrier between read & write)
        c = mm64<false>(sZ, sT2, m0, n0, lane);
        c = c * 0.25f;
        __syncthreads();
        store_tile(sZ, c, m0, n0, lane);
        __syncthreads();
    }

    // R = Z @ Q, and Q = I + Askew = B^T (skew symmetry) -> transposed-B path
    {
        v8f c = mm64<true>(sZ, sB, m0, n0, lane);
        store_tile(sT1, c, m0, n0, lane);
        __syncthreads();
    }

    // Spill R (row-major, unpadded) and mis = R^T @ li
    float* Rn = Rout + (size_t)n * DD * DD;
    for (int i = tid; i < DD * DD; i += 512)
        Rn[i] = sT1[(i >> 6) * ST + (i & 63)];
    if (tid < DD) {
        float acc = 0.0f;
#pragma unroll
        for (int i = 0; i < DD; ++i) acc += sT1[i * ST + tid] * sli[i];
        misout[n * DD + tid] = acc;
    }
}

// ---------------------------------------------------------------------------
// Phase B: tiny MLP/BN kernels (memory-trivial vs. the NS chain).
// ---------------------------------------------------------------------------

// y[n, h] = sum_d x[n, d] * W[h, d]   (Din = 64, Hout = 128)
__global__ __launch_bounds__(128) void lin_kernel(
    const float* __restrict__ x, const float* __restrict__ W,
    float* __restrict__ y)
{
    __shared__ float sx[DD];
    const int n = blockIdx.x;
    if (threadIdx.x < DD) sx[threadIdx.x] = x[(size_t)n * DD + threadIdx.x];
    __syncthreads();
    const int h = threadIdx.x;
    const float4* w4 = (const float4*)(W + (size_t)h * DD);
    float acc = 0.0f;
#pragma unroll
    for (int d = 0; d < DD / 4; ++d) {
        float4 wv = w4[d];
        acc += sx[4 * d + 0] * wv.x + sx[4 * d + 1] * wv.y +
               sx[4 * d + 2] * wv.z + sx[4 * d + 3] * wv.w;
    }
    y[(size_t)n * HH + h] = acc;
}

// Per-column training-mode BN stats -> scale = g/sqrt(var+eps), shift = b - m*scale
__global__ __launch_bounds__(256) void bn_stats_kernel(
    const float* __restrict__ h, const float* __restrict__ g,
    const float* __restrict__ b, float* __restrict__ scale,
    float* __restrict__ shift)
{
    const int col = blockIdx.x;
    float s = 0.0f, q = 0.0f;
    for (int n = threadIdx.x; n < NB; n += 256) {
        float v = h[(size_t)n * HH + col];
        s += v; q += v * v;
    }
    __shared__ float rs[256], rq[256];
    rs[threadIdx.x] = s; rq[threadIdx.x] = q;
    __syncthreads();
    for (int off = 128; off; off >>= 1) {
        if (threadIdx.x < off) {
            rs[threadIdx.x] += rs[threadIdx.x + off];
            rq[threadIdx.x] += rq[threadIdx.x + off];
        }
        __syncthreads();
    }
    if (threadIdx.x == 0) {
        float m  = rs[0] * (1.0f / NB);
        float v  = rq[0] * (1.0f / NB) - m * m;
        float sc = g[col] * rsqrtf(v + 1e-8f);
        scale[col] = sc;
        shift[col] = b[col] - m * sc;
    }
}

// s = swish(bn(h1)); fct = s @ W2^T; latent[n] = R[n] @ fct
__global__ __launch_bounds__(128) void fct_rot_kernel(
    const float* __restrict__ h1, const float* __restrict__ scale,
    const float* __restrict__ shift, const float* __restrict__ W2,
    const float* __restrict__ R, float* __restrict__ latent)
{
    __shared__ float ss[HH];
    __shared__ float sfct[DD];
    __shared__ float sR[DD * 65];
    const int n = blockIdx.x, tid = threadIdx.x;
    float v = h1[(size_t)n * HH + tid] * scale[tid] + shift[tid];
    ss[tid] = v / (1.0f + __expf(-v));
    const float* Rn = R + (size_t)n * DD * DD;
    for (int i = tid; i < DD * DD; i += 128)
        sR[(i >> 6) * 65 + (i & 63)] = Rn[i];
    __syncthreads();
    if (tid < DD) {
        float acc = 0.0f;
        const float* w = W2 + (size_t)tid * HH;
#pragma unroll
        for (int j = 0; j < HH; ++j) acc += ss[j] * w[j];
        sfct[tid] = acc;
    }
    __syncthreads();
    if (tid < DD) {
        float acc = 0.0f;
#pragma unroll
        for (int j = 0; j < DD; ++j) acc += sR[tid * 65 + j] * sfct[j];
        latent[(size_t)n * DD + tid] = acc;
    }
}

// pred = swish(bn(h2)) @ out_W2^T
__global__ __launch_bounds__(128) void out_head_kernel(
    const float* __restrict__ h2, const float* __restrict__ scale,
    const float* __restrict__ shift, const float* __restrict__ W2o,
    float* __restrict__ pred)
{
    __shared__ float ss[HH];
    const int n = blockIdx.x, tid = threadIdx.x;
    float v = h2[(size_t)n * HH + tid] * scale[tid] + shift[tid];
    ss[tid] = v / (1.0f + __expf(-v));
    __syncthreads();
    const float* w = W2o + (size_t)tid * HH;
    float acc = 0.0f;
#pragma unroll
    for (int j = 0; j < HH; ++j) acc += ss[j] * w[j];
    pred[(size_t)n * NOUT + tid] = acc;
}

// ---------------------------------------------------------------------------
// Launch. Workspace need: N*D*D + N*D + N*H + 2*H floats  (~141 MB).
// d_out layout: pred [N,128] then latent [N,64].
// ---------------------------------------------------------------------------
extern "C" void kernel_launch(void* const* d_in, const int* in_sizes, int n_in,
                              void* d_out, int out_size, void* d_ws, size_t ws_size,
                              hipStream_t stream)
{
    (void)in_sizes; (void)n_in; (void)out_size; (void)ws_size;
    const float* li   = (const float*)d_in[0];
    const float* gi   = (const float*)d_in[1];
    const float* Wup  = (const float*)d_in[2];
    const float* bup  = (const float*)d_in[3];
    const float* W1   = (const float*)d_in[4];
    const float* g1   = (const float*)d_in[5];
    const float* b1   = (const float*)d_in[6];
    const float* W2   = (const float*)d_in[7];
    const float* oW1  = (const float*)d_in[8];
    const float* og   = (const float*)d_in[9];
    const float* ob   = (const float*)d_in[10];
    const float* oW2  = (const float*)d_in[11];

    float* pred   = (float*)d_out;
    float* latent = pred + (size_t)NB * NOUT;

    float* ws    = (float*)d_ws;
    float* R     = ws;                              // N*D*D
    float* mis   = R + (size_t)NB * DD * DD;        // N*D
    float* hbuf  = mis + (size_t)NB * DD;           // N*H (reused for both MLPs)
    float* scale = hbuf + (size_t)NB * HH;          // H
    float* shift = scale + HH;                      // H

    cayley_ns_kernel<<<NB, 512, 0, stream>>>(gi, li, Wup, bup, R, mis);

    lin_kernel<<<NB, HH, 0, stream>>>(mis, W1, hbuf);
    bn_stats_kernel<<<HH, 256, 0, stream>>>(hbuf, g1, b1, scale, shift);
    fct_rot_kernel<<<NB, HH, 0, stream>>>(hbuf, scale, shift, W2, R, latent);

    lin_kernel<<<NB, HH, 0, stream>>>(latent, oW1, hbuf);
    bn_stats_kernel<<<HH, 256, 0, stream>>>(hbuf, og, ob, scale, shift);
    out_head_kernel<<<NB, HH, 0, stream>>>(hbuf, scale, shift, oW2, pred);
}